// MapEmbedding_6382321402523
// MI455X (gfx1250) — compile-verified
//
#include <hip/hip_runtime.h>
#include <hip/hip_bf16.h>

// Fused embedding gather + masked sum-pool over L=16 landmarks.
//   x: [32, 2048, 16] int32    W: [100000, 128] f32    out: [32, 2048, 128] f32
// Memory-bound: 51.2 MB table is L2-resident (192 MB L2 on MI455X); goal is
// coalesced 512B gathers from L2, SMEM index fetches, packed-f32 FMA
// accumulation, and NT-hint output stores so the streamed output does not
// evict the table from L2.

typedef float v4f __attribute__((ext_vector_type(4)));

#define EMB 128
#define LMK 16
#define WAVES_PER_BLOCK 8
#define BLOCK_THREADS (WAVES_PER_BLOCK * 32)

__global__ __launch_bounds__(BLOCK_THREADS) void
MapEmbedding_pool_kernel(const int* __restrict__ x,
                         const float* __restrict__ W,
                         float* __restrict__ out,
                         int nRows)
{
    const int lane = threadIdx.x & 31;
    // Provably wave-uniform wave id -> row id, so the 16 index loads below
    // become scalar (SMEM) loads and gather bases live in SGPRs.
    const int wv  = __builtin_amdgcn_readfirstlane((int)(threadIdx.x >> 5));
    const int row = (int)blockIdx.x * WAVES_PER_BLOCK + wv;
    if (row >= nRows) return;

    const int* __restrict__ xr = x + (size_t)row * LMK;   // uniform address

    v4f acc = (v4f){0.f, 0.f, 0.f, 0.f};

    // Fully unrolled: 16 independent global_load_b128 gathers in flight per
    // wave (LOADcnt pipelining) before the packed-FMA accumulation drains them.
#pragma unroll
    for (int l = 0; l < LMK; ++l) {
        const int idx = __builtin_amdgcn_readfirstlane(xr[l]);  // SGPR index
        // Uniform SGPR base (W + idx*512B) + divergent lane*16B offset:
        // one coalesced 512B row read per wave.
        const v4f v = ((const v4f*)(W + (size_t)idx * EMB))[lane];
        // padding_idx==0 mask; scalar select feeding v_pk_fma_f32 pairs.
        const float m = (idx != 0) ? 1.0f : 0.0f;
        acc += v * m;   // contracts to packed f32 FMA (CDNA 2x-rate path)
    }

    // Streaming output: non-temporal hint so the 33.5 MB output stream does
    // not evict the L2-resident embedding table.
    v4f* __restrict__ orow = (v4f*)(out + (size_t)row * EMB);
    __builtin_nontemporal_store(acc, orow + lane);
}

extern "C" void kernel_launch(void* const* d_in, const int* in_sizes, int n_in,
                              void* d_out, int out_size, void* d_ws, size_t ws_size,
                              hipStream_t stream)
{
    const int*   x = (const int*)  d_in[0];   // [32*2048*16] int32
    const float* W = (const float*)d_in[1];   // [100000*128] f32
    float*     out = (float*)      d_out;     // [32*2048*128] f32

    const int nRows = in_sizes[0] / LMK;      // 65536 output rows
    const int nBlocks = (nRows + WAVES_PER_BLOCK - 1) / WAVES_PER_BLOCK;

    MapEmbedding_pool_kernel<<<nBlocks, BLOCK_THREADS, 0, stream>>>(x, W, out, nRows);
}